// CustomRNN_77077483094259
// MI455X (gfx1250) — compile-verified
//
#include <hip/hip_runtime.h>
#include <hip/hip_bf16.h>

#define B_  32
#define T_  256
#define H_  128
#define KL_ 500
#define KS_ 10

typedef __attribute__((ext_vector_type(2))) float v2f;
typedef __attribute__((ext_vector_type(8))) float v8f;

// -------- phase 0a: conv feature extraction  x[b][t][0..8] --------
__global__ void features_kernel(const float* __restrict__ in1,
                                const float* __restrict__ in2,
                                const float* __restrict__ in3,
                                const float* __restrict__ wl, const float* __restrict__ bl,
                                const float* __restrict__ wsml, const float* __restrict__ bs,
                                float* __restrict__ x) {
  int idx = blockIdx.x * blockDim.x + threadIdx.x;     // over B*T*9
  if (idx >= B_ * T_ * 9) return;
  int o = idx % 9;
  int t = (idx / 9) % T_;
  int b = idx / (9 * T_);
  float acc;
  if (o < 3) {                                          // long conv, K=500
    acc = bl[o];
    const float* ip = in1 + ((size_t)b * (T_ + KL_ - 1) + t) * 3;
    const float* wp = wl + o * 3 * KL_;
    for (int k = 0; k < KL_; ++k)
      for (int i = 0; i < 3; ++i)
        acc += ip[k * 3 + i] * wp[i * KL_ + k];
  } else if (o < 6) {                                   // short conv, K=10
    int oo = o - 3;
    acc = bs[oo];
    const float* ip = in2 + ((size_t)b * (T_ + KS_ - 1) + t) * 3;
    const float* wp = wsml + oo * 3 * KS_;
    for (int k = 0; k < KS_; ++k)
      for (int i = 0; i < 3; ++i)
        acc += ip[k * 3 + i] * wp[i * KS_ + k];
  } else {                                              // passthrough input3
    acc = in3[((size_t)b * T_ + t) * 3 + (o - 6)];
  }
  x[((size_t)b * T_ + t) * 9 + o] = acc;
}

// -------- phase 0b: hb = h @ W_hh^T + b_h (constant over time); also emit h output --------
__global__ void hb_kernel(const float* __restrict__ h, const float* __restrict__ Whh,
                          const float* __restrict__ bh, float* __restrict__ hb,
                          float* __restrict__ out_h) {
  int idx = blockIdx.x * blockDim.x + threadIdx.x;     // over B*H
  if (idx >= B_ * H_) return;
  int j = idx & (H_ - 1), b = idx >> 7;
  float acc = bh[j];
  for (int k = 0; k < H_; ++k) acc += h[b * H_ + k] * Whh[j * H_ + k];
  hb[idx] = acc;
  out_h[idx] = h[idx];                                 // tuple output: h is passed through
}

// -------- persistent scan kernel: 1 workgroup, 16 waves, 256 sequential steps --------
// Adam moments for Wm live entirely in VGPRs: wave wv owns gW/Wm tiles wv*4..wv*4+3
// every timestep, so its mW/vW slices are persistent per-lane register state.
__global__ __launch_bounds__(512, 1)
void scan_kernel(const float* __restrict__ x,    // [B,T,9]
                 const float* __restrict__ hb,   // [B,H]
                 const float* __restrict__ W_ih, // [H,9]
                 const float* __restrict__ tk,   // [H,H]
                 const float* __restrict__ tv,
                 const float* __restrict__ tq,
                 const float* __restrict__ Wm0,  // [H,H]
                 const float* __restrict__ bm0,  // [H]
                 const float* __restrict__ W_ho, // [H,H]
                 const float* __restrict__ b_o,  // [H]
                 const float* __restrict__ outW, // [H]
                 const float* __restrict__ outb, // [1]
                 float* __restrict__ d_out) {    // [0..31] = final out
  __shared__ float s_Wm[H_ * H_];                // 64 KB fast weights
  __shared__ float s_out[B_ * H_];               // 16 KB out_t
  __shared__ float s_trv[B_ * H_];               // 16 KB trv (later hi)
  __shared__ float s_lg [B_ * H_];               // 16 KB lbl -> g
  __shared__ float s_qv [B_ * H_];               // 16 KB query view
  __shared__ float s_bm[H_], s_mb[H_], s_vb[H_];

  const int tid  = threadIdx.x;
  const int lane = tid & 31;
  const int wv   = tid >> 5;      // wave 0..15
  const int hi   = lane >> 4;     // lane half (selects K+2 / M+8)
  const int lo   = lane & 15;

  for (int i = tid; i < H_ * H_; i += 512) s_Wm[i] = Wm0[i];
  for (int i = tid; i < H_; i += 512) { s_bm[i] = bm0[i]; s_mb[i] = 0.f; s_vb[i] = 0.f; }
  __syncthreads();

  const float lr = 0.01f, bt1 = 0.9f, bt2 = 0.999f, eps = 1e-8f;
  const float c0 = 2.0f / (float)(B_ * H_);
  float b1p = 1.f, b2p = 1.f;

  // register-resident Adam moments for this wave's 4 Wm tiles
  v8f r_mW[4] = {v8f{}, v8f{}, v8f{}, v8f{}};
  v8f r_vW[4] = {v8f{}, v8f{}, v8f{}, v8f{}};

  const int mi = wv >> 3;         // tile row of [32,H] outputs (0..1)
  const int ni = wv & 7;          // tile col (0..7)
  const int m  = mi * 16 + lo;    // A-row for this lane
  const int n  = ni * 16 + lo;    // B-col for this lane

  for (int t = 0; t < T_; ++t) {
    b1p *= bt1; b2p *= bt2;
    const float bc1 = 1.f - b1p, bc2 = 1.f - b2p;

    // ---- out_t = x_t @ W_ih^T + hb   (K=9, VALU) ----
    for (int i = tid; i < B_ * H_; i += 512) {
      int bb = i >> 7, j = i & (H_ - 1);
      const float* xp = x + ((size_t)bb * T_ + t) * 9;
      const float* wp = W_ih + j * 9;
      float acc = hb[i];
#pragma unroll
      for (int k = 0; k < 9; ++k) acc += xp[k] * wp[k];
      s_out[i] = acc;
    }
    __syncthreads();

    // ---- trv / lbl / qv = out_t @ {tk,tv,tq}   (WMMA f32 16x16x4, K=128) ----
    {
      v8f a0 = {}, a1 = {}, a2 = {};
      for (int k = 0; k < H_; k += 4) {
        int kk = k + 2 * hi;
        v2f av; av.x = s_out[m * H_ + kk]; av.y = s_out[m * H_ + kk + 1];
        v2f bk; bk.x = tk[kk * H_ + n];  bk.y = tk[(kk + 1) * H_ + n];
        v2f bv; bv.x = tv[kk * H_ + n];  bv.y = tv[(kk + 1) * H_ + n];
        v2f bq; bq.x = tq[kk * H_ + n];  bq.y = tq[(kk + 1) * H_ + n];
        a0 = __builtin_amdgcn_wmma_f32_16x16x4_f32(false, av, false, bk, (short)0, a0, false, false);
        a1 = __builtin_amdgcn_wmma_f32_16x16x4_f32(false, av, false, bv, (short)0, a1, false, false);
        a2 = __builtin_amdgcn_wmma_f32_16x16x4_f32(false, av, false, bq, (short)0, a2, false, false);
      }
#pragma unroll
      for (int r = 0; r < 8; ++r) {
        int row = mi * 16 + r + 8 * hi, col = ni * 16 + lo;
        s_trv[row * H_ + col] = a0[r];
        s_lg [row * H_ + col] = a1[r];
        s_qv [row * H_ + col] = a2[r];
      }
    }
    __syncthreads();

    // ---- pred = trv @ Wm^T + bm ; g = c0*(pred - lbl) in place into s_lg ----
    {
      v8f acc = {};
      for (int k = 0; k < H_; k += 4) {
        int kk = k + 2 * hi;
        v2f av; av.x = s_trv[m * H_ + kk]; av.y = s_trv[m * H_ + kk + 1];
        v2f bv; bv.x = s_Wm[n * H_ + kk];  bv.y = s_Wm[n * H_ + kk + 1]; // Wm^T
        acc = __builtin_amdgcn_wmma_f32_16x16x4_f32(false, av, false, bv, (short)0, acc, false, false);
      }
#pragma unroll
      for (int r = 0; r < 8; ++r) {
        int row = mi * 16 + r + 8 * hi, col = ni * 16 + lo;
        float pred = acc[r] + s_bm[col];
        s_lg[row * H_ + col] = c0 * (pred - s_lg[row * H_ + col]);
      }
    }
    __syncthreads();

    // ---- gW = g^T @ trv (K=32) fused with register-resident Adam on Wm ----
#pragma unroll
    for (int tt4 = 0; tt4 < 4; ++tt4) {
      int tile = wv * 4 + tt4;
      int gmi = tile >> 3, gni = tile & 7;
      v8f acc = {};
      for (int k = 0; k < B_; k += 4) {
        int kk = k + 2 * hi;
        v2f av; av.x = s_lg [kk * H_ + gmi * 16 + lo]; av.y = s_lg [(kk + 1) * H_ + gmi * 16 + lo];
        v2f bv; bv.x = s_trv[kk * H_ + gni * 16 + lo]; bv.y = s_trv[(kk + 1) * H_ + gni * 16 + lo];
        acc = __builtin_amdgcn_wmma_f32_16x16x4_f32(false, av, false, bv, (short)0, acc, false, false);
      }
#pragma unroll
      for (int r = 0; r < 8; ++r) {
        int row = gmi * 16 + r + 8 * hi, col = gni * 16 + lo;
        int idx = row * H_ + col;
        float g  = acc[r];
        float mm = bt1 * r_mW[tt4][r] + (1.f - bt1) * g;
        float vv = bt2 * r_vW[tt4][r] + (1.f - bt2) * g * g;
        r_mW[tt4][r] = mm; r_vW[tt4][r] = vv;
        s_Wm[idx] -= lr * (mm / bc1) / (sqrtf(vv / bc2) + eps);
      }
    }
    if (tid < H_) {                                   // bm Adam update (LDS moments)
      float gb = 0.f;
      for (int bb = 0; bb < B_; ++bb) gb += s_lg[bb * H_ + tid];
      float mm = bt1 * s_mb[tid] + (1.f - bt1) * gb;
      float vv = bt2 * s_vb[tid] + (1.f - bt2) * gb * gb;
      s_mb[tid] = mm; s_vb[tid] = vv;
      s_bm[tid] -= lr * (mm / bc1) / (sqrtf(vv / bc2) + eps);
    }
    __syncthreads();

    // ---- hi = qv @ Wm_new^T + bm_new  -> reuse s_trv ----
    {
      v8f acc = {};
      for (int k = 0; k < H_; k += 4) {
        int kk = k + 2 * hi;
        v2f av; av.x = s_qv[m * H_ + kk]; av.y = s_qv[m * H_ + kk + 1];
        v2f bv; bv.x = s_Wm[n * H_ + kk]; bv.y = s_Wm[n * H_ + kk + 1];
        acc = __builtin_amdgcn_wmma_f32_16x16x4_f32(false, av, false, bv, (short)0, acc, false, false);
      }
#pragma unroll
      for (int r = 0; r < 8; ++r) {
        int row = mi * 16 + r + 8 * hi, col = ni * 16 + lo;
        s_trv[row * H_ + col] = acc[r] + s_bm[col];
      }
    }
    __syncthreads();
  }

  // ---- only last timestep feeds output: out = (hi @ W_ho^T + b_o) @ out_W^T + out_b ----
  if (tid < B_) {
    float o = 0.f;
    for (int nn = 0; nn < H_; ++nn) {
      float y = b_o[nn];
      for (int k = 0; k < H_; ++k) y += s_trv[tid * H_ + k] * W_ho[nn * H_ + k];
      o += y * outW[nn];
    }
    d_out[tid] = o + outb[0];
  }
}

extern "C" void kernel_launch(void* const* d_in, const int* in_sizes, int n_in,
                              void* d_out, int out_size, void* d_ws, size_t ws_size,
                              hipStream_t stream) {
  const float* in1  = (const float*)d_in[0];
  const float* in2  = (const float*)d_in[1];
  const float* in3  = (const float*)d_in[2];
  const float* h    = (const float*)d_in[3];
  const float* W_ih = (const float*)d_in[4];
  const float* W_hh = (const float*)d_in[5];
  const float* b_h  = (const float*)d_in[6];
  const float* W_ho = (const float*)d_in[7];
  const float* b_o  = (const float*)d_in[8];
  const float* wl   = (const float*)d_in[9];
  const float* bl   = (const float*)d_in[10];
  const float* wsml = (const float*)d_in[11];
  const float* bs   = (const float*)d_in[12];
  const float* tk   = (const float*)d_in[13];
  const float* tv   = (const float*)d_in[14];
  const float* tq   = (const float*)d_in[15];
  const float* Wm0  = (const float*)d_in[16];
  const float* bm0  = (const float*)d_in[17];
  const float* oW   = (const float*)d_in[18];
  const float* ob   = (const float*)d_in[19];
  float* out = (float*)d_out;

  float* ws = (float*)d_ws;
  float* x  = ws;               // 73728 floats: [B,T,9]
  float* hb = ws + 73728;       // 4096

  features_kernel<<<(B_ * T_ * 9 + 255) / 256, 256, 0, stream>>>(in1, in2, in3, wl, bl, wsml, bs, x);
  hb_kernel<<<(B_ * H_ + 255) / 256, 256, 0, stream>>>(h, W_hh, b_h, hb, out + 32);
  scan_kernel<<<1, 512, 0, stream>>>(x, hb, W_ih, tk, tv, tq, Wm0, bm0,
                                     W_ho, b_o, oW, ob, out);
}